// SpikeTime_33681133535236
// MI455X (gfx1250) — compile-verified
//
#include <hip/hip_runtime.h>
#include <hip/hip_bf16.h>
#include <stdint.h>

// Problem geometry (fixed by the reference).
#define T_STEPS 128
#define BATCH   128
#define N_OUT   4096
#define BN      (BATCH * N_OUT)   // 524288 columns
#define VEC     4                 // columns per thread (one b128 per lane per plane)
#define BLOCK   256               // 8 wave32 waves per block
#define WAVES_PER_BLOCK (BLOCK / 32)

// ---- CDNA5 async global->LDS + asynccnt plumbing --------------------------

// Builtin signature (from hipcc diagnostic): params are pointers to GNU-vector
// 'int __vector(4)' in AS1 (global / __device__) and AS3 (LDS) respectively.
typedef int v4i __attribute__((vector_size(16)));
typedef __attribute__((address_space(1))) v4i as1_v4i;
typedef __attribute__((address_space(3))) v4i as3_v4i;
typedef __attribute__((address_space(3))) char as3_char;

#if __has_builtin(__builtin_amdgcn_global_load_async_to_lds_b128)
#define HAVE_ASYNC_LDS_BUILTIN 1
#endif
#if __has_builtin(__builtin_amdgcn_s_wait_asynccnt)
#define HAVE_WAIT_ASYNC_BUILTIN 1
#endif

template <int N>
__device__ __forceinline__ void wait_asynccnt_le() {
#if defined(HAVE_WAIT_ASYNC_BUILTIN)
    __builtin_amdgcn_s_wait_asynccnt(N);
#else
    asm volatile("s_wait_asynccnt %0" :: "n"(N) : "memory");
#endif
    asm volatile("" ::: "memory");   // keep the LDS read below the wait
}

// Async DMA 16 bytes per lane: global (gp) -> LDS (lp). Tracked by ASYNCcnt.
__device__ __forceinline__ void async_load_b128_to_lds(const float* gp, float* lp) {
#if defined(HAVE_ASYNC_LDS_BUILTIN)
    __builtin_amdgcn_global_load_async_to_lds_b128(
        (as1_v4i*)gp, (as3_v4i*)lp, /*imm offset*/0, /*cpol*/0);
#else
    unsigned lds_off = (unsigned)(uintptr_t)(as3_char*)lp;     // LDS byte offset
    unsigned long long ga = (unsigned long long)(uintptr_t)gp; // 64-bit vaddr
    asm volatile("global_load_async_to_lds_b128 %0, %1, off"
                 :: "v"(lds_off), "v"(ga) : "memory");
#endif
}

// ---------------------------------------------------------------------------
// First-spike-time scan with wave-uniform early exit and a depth-2 async
// global->LDS pipeline. spk layout: [T, B*N], column-contiguous per plane.
// out[0 .. BN)        = first spike step (or T-1 if none), as float
// out[BN .. 2*BN)     = wrap-corrected targets, as float
// ---------------------------------------------------------------------------
__global__ __launch_bounds__(BLOCK)
void SpikeTime_first_spike_kernel(const float* __restrict__ spk,
                                  const int*   __restrict__ tgt,
                                  float*       __restrict__ out) {
    __shared__ float stage[WAVES_PER_BLOCK][2][32 * VEC];   // 8 KB, per-wave ping-pong

    const int tid  = threadIdx.x;
    const int wave = tid >> 5;
    const int lane = tid & 31;
    const int col4 = (blockIdx.x * BLOCK + tid) * VEC;      // first of my 4 columns

    float* slot0 = &stage[wave][0][lane * VEC];
    float* slot1 = &stage[wave][1][lane * VEC];

    // Prime the pipeline with plane t = 0.
    async_load_b128_to_lds(spk + col4, slot0);

    int  r0 = T_STEPS - 1, r1 = T_STEPS - 1, r2 = T_STEPS - 1, r3 = T_STEPS - 1;
    bool f0 = false, f1 = false, f2 = false, f3 = false;

    for (int t = 0; t < T_STEPS; ++t) {
        float* cur = (t & 1) ? slot1 : slot0;
        if (t + 1 < T_STEPS) {
            float* nxt = ((t + 1) & 1) ? slot1 : slot0;
            async_load_b128_to_lds(spk + (size_t)(t + 1) * BN + col4, nxt);
            wait_asynccnt_le<1>();   // retire plane t (async loads complete in order)
        } else {
            wait_asynccnt_le<0>();
        }

        const float4 v = *reinterpret_cast<const float4*>(cur);  // ds_load_b128
        if (!f0 && v.x != 0.0f) { r0 = t; f0 = true; }
        if (!f1 && v.y != 0.0f) { r1 = t; f1 = true; }
        if (!f2 && v.z != 0.0f) { r2 = t; f2 = true; }
        if (!f3 && v.w != 0.0f) { r3 = t; f3 = true; }

        // Wave-uniform early exit: E[max first-spike over 32 lanes * 4 cols] ~ 9
        // steps, so this drops HBM traffic ~10x below the full 256 MiB scan.
        if (__all(f0 && f1 && f2 && f3)) break;
    }
    wait_asynccnt_le<0>();  // drain the possibly-unconsumed prefetch

    float4 res;
    res.x = (float)r0; res.y = (float)r1; res.z = (float)r2; res.w = (float)r3;
    *reinterpret_cast<float4*>(out + col4) = res;

    // Second output: wrap-corrected targets (t < 0 -> T + t) as float.
    const int4 tg = *reinterpret_cast<const int4*>(tgt + col4);
    float4 wt;
    wt.x = (float)(tg.x < 0 ? T_STEPS + tg.x : tg.x);
    wt.y = (float)(tg.y < 0 ? T_STEPS + tg.y : tg.y);
    wt.z = (float)(tg.z < 0 ? T_STEPS + tg.z : tg.z);
    wt.w = (float)(tg.w < 0 ? T_STEPS + tg.w : tg.w);
    *reinterpret_cast<float4*>(out + BN + col4) = wt;
}

extern "C" void kernel_launch(void* const* d_in, const int* in_sizes, int n_in,
                              void* d_out, int out_size, void* d_ws, size_t ws_size,
                              hipStream_t stream) {
    (void)in_sizes; (void)n_in; (void)out_size; (void)d_ws; (void)ws_size;
    const float* spk = (const float*)d_in[0];   // [T, B, N] float32
    const int*   tgt = (const int*)d_in[1];     // [B, N] integer
    float*       out = (float*)d_out;           // [2, B, N] float32 (concat outputs)

    const int threads = BLOCK;
    const int blocks  = BN / (BLOCK * VEC);     // 524288 / 1024 = 512
    SpikeTime_first_spike_kernel<<<blocks, threads, 0, stream>>>(spk, tgt, out);
}